// GCNLayer_5944234738328
// MI455X (gfx1250) — compile-verified
//
#include <hip/hip_runtime.h>

typedef __attribute__((ext_vector_type(2))) float v2f;
typedef __attribute__((ext_vector_type(8))) float v8f;

#define D_FEAT 64

// out[16t : 16t+16, 0:64] = sum over this tile's edge range of vals[e] * embeds[cols[e]]
// C(16x16) += A(16x4) x B(4x16) FP32 WMMAs, 4 feature blocks of 16 columns each.
// rows sorted -> each 16-row tile owns a contiguous, disjoint edge range -> no atomics.
__global__ __launch_bounds__(256) void spmm_wmma_f32(
    const int*   __restrict__ rows,
    const int*   __restrict__ cols,
    const float* __restrict__ vals,
    const float* __restrict__ embeds,
    float*       __restrict__ out,
    int n_edges, int n_nodes, int n_tiles)
{
    const int lane = threadIdx.x & 31;
    const int wave = threadIdx.x >> 5;
    const int tile = blockIdx.x * (blockDim.x >> 5) + wave;
    if (tile >= n_tiles) return;             // wave-uniform exit: EXEC stays all-1s

    const int row_base = tile * 16;
    const int row_lim  = row_base + 16;

    // Binary search (uniform across wave): edge range [e_begin, e_end) for this tile.
    int lo = 0, hi = n_edges;
    while (lo < hi) { int mid = (lo + hi) >> 1; if (rows[mid] < row_base) lo = mid + 1; else hi = mid; }
    const int e_begin = lo;
    hi = n_edges;
    while (lo < hi) { int mid = (lo + hi) >> 1; if (rows[mid] < row_lim) lo = mid + 1; else hi = mid; }
    const int e_end = lo;

    const int half   = lane >> 4;     // 0: handles K=0,1   1: handles K=2,3
    const int m      = lane & 15;     // A: row M. B/C: column N.
    const int target = row_base + m;  // row this lane one-hots in A

    v8f c0 = {}, c1 = {}, c2 = {}, c3 = {};

    // ---- main loop: full groups of 4 edges, branch-free ----
    const int e_full_end = e_begin + ((e_end - e_begin) & ~3);

#pragma unroll 2
    for (int e = e_begin; e < e_full_end; e += 4) {
        const int ea = e + 2 * half;      // K = 2*half
        const int eb = ea + 1;            // K = 2*half + 1

        const int   ra = rows[ea];
        const int   rb = rows[eb];
        const float va = vals[ea];
        const float vb = vals[eb];
        const int   ia = cols[ea] * D_FEAT + m;   // 32-bit gather index (fits: 50k*64)
        const int   ib = cols[eb] * D_FEAT + m;

        v2f A;
        A.x = (ra == target) ? va : 0.0f;
        A.y = (rb == target) ? vb : 0.0f;

        v2f B0, B1, B2, B3;
        B0.x = embeds[ia];       B0.y = embeds[ib];
        B1.x = embeds[ia + 16];  B1.y = embeds[ib + 16];
        B2.x = embeds[ia + 32];  B2.y = embeds[ib + 32];
        B3.x = embeds[ia + 48];  B3.y = embeds[ib + 48];

        c0 = __builtin_amdgcn_wmma_f32_16x16x4_f32(false, A, false, B0, (short)0, c0, false, false);
        c1 = __builtin_amdgcn_wmma_f32_16x16x4_f32(false, A, false, B1, (short)0, c1, false, false);
        c2 = __builtin_amdgcn_wmma_f32_16x16x4_f32(false, A, false, B2, (short)0, c2, false, false);
        c3 = __builtin_amdgcn_wmma_f32_16x16x4_f32(false, A, false, B3, (short)0, c3, false, false);
    }

    // ---- tail: one masked group (wave-uniform entry -> EXEC full at the WMMAs) ----
    if (e_full_end < e_end) {
        const int ea = e_full_end + 2 * half;
        const int eb = ea + 1;

        float a0 = 0.0f, a1 = 0.0f;
        int   ia = m, ib = m;
        if (ea < e_end) {
            a0 = (rows[ea] == target) ? vals[ea] : 0.0f;
            ia = cols[ea] * D_FEAT + m;
        }
        if (eb < e_end) {
            a1 = (rows[eb] == target) ? vals[eb] : 0.0f;
            ib = cols[eb] * D_FEAT + m;
        }
        v2f A; A.x = a0; A.y = a1;

        v2f B0, B1, B2, B3;
        B0.x = embeds[ia];       B0.y = embeds[ib];
        B1.x = embeds[ia + 16];  B1.y = embeds[ib + 16];
        B2.x = embeds[ia + 32];  B2.y = embeds[ib + 32];
        B3.x = embeds[ia + 48];  B3.y = embeds[ib + 48];

        c0 = __builtin_amdgcn_wmma_f32_16x16x4_f32(false, A, false, B0, (short)0, c0, false, false);
        c1 = __builtin_amdgcn_wmma_f32_16x16x4_f32(false, A, false, B1, (short)0, c1, false, false);
        c2 = __builtin_amdgcn_wmma_f32_16x16x4_f32(false, A, false, B2, (short)0, c2, false, false);
        c3 = __builtin_amdgcn_wmma_f32_16x16x4_f32(false, A, false, B3, (short)0, c3, false, false);
    }

    // Store C tile: VGPR v -> row M = v + 8*half, column N = m (+ feature-block base).
#pragma unroll
    for (int v = 0; v < 8; ++v) {
        const int row = row_base + v + 8 * half;
        if (row < n_nodes) {
            float* o = out + (long)row * D_FEAT + m;
            o[0]  = c0[v];
            o[16] = c1[v];
            o[32] = c2[v];
            o[48] = c3[v];
        }
    }
}

extern "C" void kernel_launch(void* const* d_in, const int* in_sizes, int n_in,
                              void* d_out, int out_size, void* d_ws, size_t ws_size,
                              hipStream_t stream) {
    const int*   rows   = (const int*)d_in[0];
    const int*   cols   = (const int*)d_in[1];
    const float* vals   = (const float*)d_in[2];
    const float* embeds = (const float*)d_in[3];
    float*       out    = (float*)d_out;

    const int n_edges = in_sizes[0];
    const int n_nodes = in_sizes[3] / D_FEAT;
    const int n_tiles = (n_nodes + 15) / 16;

    const int threads = 256;                      // 8 waves per block (wave32)
    const int waves_per_block = threads / 32;
    const int blocks = (n_tiles + waves_per_block - 1) / waves_per_block;

    spmm_wmma_f32<<<blocks, threads, 0, stream>>>(rows, cols, vals, embeds, out,
                                                  n_edges, n_nodes, n_tiles);
}